// CombinedSolver_36472862277866
// MI455X (gfx1250) — compile-verified
//
#include <hip/hip_runtime.h>
#include <math.h>

// Problem constants (fixed by the reference): Tp1=513, B=256, d=32
#define TDIM 512
#define BDIM 256
#define DDIM 32
#define ROWS 128          // (t,b)-rows per phaseA block -> one t per block
#define LPAD 36           // LDS row stride (dwords): 16B-aligned, bank-spread

typedef float v2f __attribute__((ext_vector_type(2)));
typedef float v8f __attribute__((ext_vector_type(8)));

// D(16x16) = A(16x4) x B(4x16) + C  , f32, wave32
__device__ __forceinline__ v8f wmma4(v2f a, v2f b, v8f c) {
  return __builtin_amdgcn_wmma_f32_16x16x4_f32(false, a, false, b, (short)0, c,
                                               false, false);
}

// ---------------------------------------------------------------------------
// Kernel 0: sigma_inv via Gauss-Jordan (32x32, LDS) + sgg[b,i] = (g[b] @ sigma)[i]
// ---------------------------------------------------------------------------
__global__ __launch_bounds__(1024) void k0_prepare(
    const float* __restrict__ sigma, const float* __restrict__ nabla_g,
    float* __restrict__ ws_si, float* __restrict__ ws_sgg) {
  __shared__ float A[32 * 33];
  __shared__ float Iv[32 * 33];
  __shared__ float S[32 * 33];
  const int tid = threadIdx.x;
  const int i = tid >> 5, j = tid & 31;
  const float sij = sigma[i * 32 + j];
  A[i * 33 + j] = sij;
  S[i * 33 + j] = sij;
  Iv[i * 33 + j] = (i == j) ? 1.0f : 0.0f;
  __syncthreads();
  for (int k = 0; k < 32; ++k) {
    const float akk = A[k * 33 + k];
    const float aik = A[i * 33 + k];
    const float akj = A[k * 33 + j];
    const float ikj = Iv[k * 33 + j];
    __syncthreads();
    if (i == k) {
      A[k * 33 + j] = akj / akk;
      Iv[k * 33 + j] = ikj / akk;
    } else {
      const float f = aik / akk;
      A[i * 33 + j] -= f * akj;
      Iv[i * 33 + j] -= f * ikj;
    }
    __syncthreads();
  }
  ws_si[i * 32 + j] = Iv[i * 33 + j];
  // sgg[b][c] = sum_j g[b][j] * sigma[j][c]
  for (int o = tid; o < BDIM * DDIM; o += 1024) {
    const int b = o >> 5, c = o & 31;
    float s = 0.0f;
#pragma unroll
    for (int jj = 0; jj < 32; ++jj) s += nabla_g[b * 32 + jj] * S[jj * 33 + c];
    ws_sgg[o] = s;
  }
}

// ---------------------------------------------------------------------------
// Phase A: per 128 rows (one t):
//   preV = sqrt(lmbd*dt)*noises + dt*controls          (LDS)
//   v    = preV @ sigma_inv                            (WMMA f32 16x16x4)
//   aC[r][i] = dt*nabla_f[r][i] - dot(nabla_b[r][i][:], v[r][:])   (streaming)
//   bc   = aC @ sigma                                  (WMMA) -> global ws
// ---------------------------------------------------------------------------
__global__ __launch_bounds__(256) void phaseA(
    const float* __restrict__ noises, const float* __restrict__ controls,
    const float* __restrict__ nabla_f, const float* __restrict__ nabla_b,
    const float* __restrict__ ts, const float* __restrict__ ws_si,
    const float* __restrict__ sigma, float* __restrict__ bc) {
  __shared__ float sPreV[ROWS * LPAD];  // aliased as aComb after barrier 2
  __shared__ float sV[ROWS * LPAD];
  __shared__ float sSi[DDIM * LPAD];
  __shared__ float sSg[DDIM * LPAD];

  const int tid = threadIdx.x;
  const int rowBase = blockIdx.x * ROWS;          // flattened (t,b) row
  const int t = rowBase / BDIM;                   // constant per block
  const float dt = ts[t + 1] - ts[t];
  const float s1 = sqrtf(1.0f /*LMBD*/ * dt);

  for (int o = tid; o < DDIM * DDIM; o += 256) {
    const int r = o >> 5, c = o & 31;
    sSi[r * LPAD + c] = ws_si[o];
    sSg[r * LPAD + c] = sigma[o];
  }
  const float* nP = noises + (size_t)rowBase * DDIM;
  const float* cP = controls + (size_t)rowBase * DDIM;
  for (int o = tid; o < ROWS * DDIM; o += 256) {
    const int r = o >> 5, c = o & 31;
    sPreV[r * LPAD + c] = s1 * nP[o] + dt * cP[o];
  }
  __syncthreads();

  const int wv = tid >> 5;        // wave id 0..7, owns 16 rows for WMMA
  const int lane = tid & 31;
  const int ml = lane & 15;
  const int hi = lane >> 4;
  const int rb = wv * 16;

  // ---- GEMM1: sV = sPreV @ sigma_inv ----
#pragma unroll
  for (int h = 0; h < 2; ++h) {
    v8f acc = {};
#pragma unroll
    for (int kk = 0; kk < 8; ++kk) {
      const int k0 = kk * 4 + hi * 2;
      v2f a, b;
      a.x = sPreV[(rb + ml) * LPAD + k0];
      a.y = sPreV[(rb + ml) * LPAD + k0 + 1];
      b.x = sSi[k0 * LPAD + h * 16 + ml];
      b.y = sSi[(k0 + 1) * LPAD + h * 16 + ml];
      acc = wmma4(a, b, acc);
    }
#pragma unroll
    for (int r = 0; r < 8; ++r)
      sV[(rb + r + 8 * hi) * LPAD + h * 16 + ml] = acc[r];
  }
  __syncthreads();

  // ---- streaming matvec over nabla_b (bandwidth-dominant) ----
  const float* nbB = nabla_b + (size_t)rowBase * (DDIM * DDIM);
  const float* nfB = nabla_f + (size_t)rowBase * DDIM;
#pragma unroll 2
  for (int kc = 0; kc < 16; ++kc) {
    const int r = wv + 8 * kc;   // wave-uniform row
    float4 vv[8];
#pragma unroll
    for (int q = 0; q < 8; ++q) vv[q] = *(const float4*)&sV[r * LPAD + 4 * q];
    const float4* nb4 = (const float4*)(nbB + ((size_t)r * DDIM + lane) * DDIM);
    float dotv = 0.0f;
#pragma unroll
    for (int q = 0; q < 8; ++q) {
      const float4 m = nb4[q];
      dotv += m.x * vv[q].x + m.y * vv[q].y + m.z * vv[q].z + m.w * vv[q].w;
    }
    sPreV[r * LPAD + lane] = dt * nfB[r * DDIM + lane] - dotv;  // aComb
  }
  __syncthreads();

  // ---- GEMM2: bc = aComb @ sigma ----
#pragma unroll
  for (int h = 0; h < 2; ++h) {
    v8f acc = {};
#pragma unroll
    for (int kk = 0; kk < 8; ++kk) {
      const int k0 = kk * 4 + hi * 2;
      v2f a, b;
      a.x = sPreV[(rb + ml) * LPAD + k0];
      a.y = sPreV[(rb + ml) * LPAD + k0 + 1];
      b.x = sSg[k0 * LPAD + h * 16 + ml];
      b.y = sSg[(k0 + 1) * LPAD + h * 16 + ml];
      acc = wmma4(a, b, acc);
    }
#pragma unroll
    for (int r = 0; r < 8; ++r)
      bc[(size_t)(rowBase + rb + r + 8 * hi) * DDIM + h * 16 + ml] = acc[r];
  }
}

// ---------------------------------------------------------------------------
// Phase C: backward suffix scan over t + weighted SSE. gid <-> (b,i) so every
// load is contiguous across the block. Partials per block (fixed-order sum
// later for determinism).
// ---------------------------------------------------------------------------
__global__ __launch_bounds__(256) void phaseC(
    const float* __restrict__ cl, const float* __restrict__ lw,
    const float* __restrict__ bc, const float* __restrict__ sgg,
    float* __restrict__ partial) {
  const int tid = threadIdx.x;
  const int gid = blockIdx.x * 256 + tid;  // 0..8191 = b*32+i
  const int b = gid >> 5;
  const float sggv = sgg[gid];
  float suf = 0.0f, acc = 0.0f;
#pragma unroll 4
  for (int t = TDIM; t >= 0; --t) {
    const float c = cl[(size_t)t * (BDIM * DDIM) + gid];
    const float e = c + sggv + suf;   // cl - control_target
    acc += e * e;
    if (t > 0) suf += bc[(size_t)(t - 1) * (BDIM * DDIM) + gid];
  }
  acc *= __expf(lw[b]);
  __shared__ float red[256];
  red[tid] = acc;
  __syncthreads();
  for (int s = 128; s > 0; s >>= 1) {
    if (tid < s) red[tid] += red[tid + s];
    __syncthreads();
  }
  if (tid == 0) partial[blockIdx.x] = red[0];
}

__global__ void finalReduce(const float* __restrict__ partial,
                            float* __restrict__ out) {
  if (threadIdx.x == 0) {
    float s = 0.0f;
    for (int i = 0; i < 32; ++i) s += partial[i];
    out[0] = s * (1.0f / (513.0f * 256.0f));
  }
}

extern "C" void kernel_launch(void* const* d_in, const int* in_sizes, int n_in,
                              void* d_out, int out_size, void* d_ws,
                              size_t ws_size, hipStream_t stream) {
  (void)in_sizes; (void)n_in; (void)out_size; (void)ws_size;
  // setup_inputs order: states, noises, controls, nabla_f, nabla_b, nabla_g,
  //                     control_learned, log_weight, sigma, ts
  const float* noises   = (const float*)d_in[1];
  const float* controls = (const float*)d_in[2];
  const float* nabla_f  = (const float*)d_in[3];
  const float* nabla_b  = (const float*)d_in[4];
  const float* nabla_g  = (const float*)d_in[5];
  const float* cl       = (const float*)d_in[6];
  const float* lw       = (const float*)d_in[7];
  const float* sigma    = (const float*)d_in[8];
  const float* ts       = (const float*)d_in[9];
  float* out = (float*)d_out;

  float* ws      = (float*)d_ws;
  float* ws_bc   = ws;                       // 512*256*32 = 4194304 floats
  float* ws_si   = ws_bc + 4194304;          // 1024 floats
  float* ws_sgg  = ws_si + 1024;             // 8192 floats
  float* ws_part = ws_sgg + 8192;            // 32 floats

  k0_prepare<<<1, 1024, 0, stream>>>(sigma, nabla_g, ws_si, ws_sgg);
  phaseA<<<(TDIM * BDIM) / ROWS, 256, 0, stream>>>(noises, controls, nabla_f,
                                                   nabla_b, ts, ws_si, sigma,
                                                   ws_bc);
  phaseC<<<32, 256, 0, stream>>>(cl, lw, ws_bc, ws_sgg, ws_part);
  finalReduce<<<1, 32, 0, stream>>>(ws_part, out);
}